// DemandRouter_28132035789119
// MI455X (gfx1250) — compile-verified
//
#include <hip/hip_runtime.h>
#include <stdint.h>

// DemandRouter for MI455X (gfx1250, wave32).
// f32 WMMA (v_wmma_f32_16x16x4_f32) for both projection and similarity GEMMs;
// A-operands staged per-workgroup into LDS via async global->LDS (ASYNCcnt),
// since all 8 waves of a block share the same A fragment (8x reuse).
// B-operands (W rows / k rows) have no cross-wave reuse -> direct L2 streams.

typedef __attribute__((ext_vector_type(2))) float v2f;
typedef __attribute__((ext_vector_type(8))) float v8f;
typedef __attribute__((ext_vector_type(4))) int   v4i;

namespace {
constexpr int D    = 1024;   // feature dim
constexpr int KQ   = 128;    // q/k dim
constexpr int TT   = 2048;   // tokens per batch
constexpr int NB   = 8;      // batches
constexpr int TOPK = 8;
}

// ---- async global->LDS staging (gfx1250 ASYNCcnt path, guarded) -----------
#if defined(__AMDGCN__) && __has_builtin(__builtin_amdgcn_global_load_async_to_lds_b128)
#define HAVE_ASYNC_LDS 1
#else
#define HAVE_ASYNC_LDS 0
#endif

// builtin signature (from hipcc diagnostic): param0 = global-AS pointer to
// int __vector(4); param1 = LDS-AS pointer; then (imm offset, imm cpol).
typedef __attribute__((address_space(1))) v4i gbl_v4i;
typedef __attribute__((address_space(3))) v4i lds_v4i;

__device__ __forceinline__ void async_copy_b128(const float* g, float* l) {
#if HAVE_ASYNC_LDS
  __builtin_amdgcn_global_load_async_to_lds_b128(
      (gbl_v4i*)(uintptr_t)g, (lds_v4i*)(uintptr_t)l, 0, 0);
#else
  *reinterpret_cast<float4*>(l) = *reinterpret_cast<const float4*>(g);
#endif
}

__device__ __forceinline__ void async_join() {
#if HAVE_ASYNC_LDS
#if __has_builtin(__builtin_amdgcn_s_wait_asynccnt)
  __builtin_amdgcn_s_wait_asynccnt(0);
#else
  asm volatile("s_wait_asynccnt 0x0" ::: "memory");
#endif
#endif
}

// ---------------------------------------------------------------------------
// Kernel 1: fused Q and K projection.  grid = B*T/16 blocks, 256 threads.
// x tile [16 x 1024] (64 KB) async-staged to LDS with a 4-float row stagger
// ((c + 4r) & 1023): 16 A-lanes hit 16 distinct banks, b128/b64 alignment kept.
// Wave w computes output columns [16w, 16w+16); Wq/Wk rows stream from L2.
//
// f32 WMMA layouts (ISA 7.12.2):
//   A 16x4 : lane l -> row M = l&15 ; V0 = K(2*(l>>4)), V1 = K(2*(l>>4)+1)
//   B 4x16 : lane l -> col N = l&15 ; V0 = K(2*(l>>4)), V1 = K(2*(l>>4)+1)
//   C 16x16: VGPR r -> row M = r + 8*(l>>4), col N = l&15
// ---------------------------------------------------------------------------
__global__ __launch_bounds__(256) void qk_proj_kernel(
    const float* __restrict__ x,  const float* __restrict__ Wq,
    const float* __restrict__ bq, const float* __restrict__ Wk,
    const float* __restrict__ bk, float* __restrict__ qout,
    float* __restrict__ kout)
{
  __shared__ float xt[16 * 1024];           // 64 KB, row-staggered
  const int tid = threadIdx.x;
  const int bt0 = blockIdx.x * 16;          // flat token index of tile start

  for (int i = tid; i < 16 * 1024 / 4; i += 256) {
    const int e = i * 4;
    const int r = e >> 10, c = e & 1023;
    async_copy_b128(&x[(size_t)(bt0 + r) * D + c],
                    &xt[r * 1024 + ((c + r * 4) & 1023)]);
  }
  async_join();
  __syncthreads();

  const int w   = tid >> 5;                 // wave id -> n-tile
  const int l   = tid & 31;                 // lane
  const int m   = l & 15;                   // A row / B col within tile
  const int kof = (l >> 4) * 2;             // lane K offset inside 4-wide step

  const float* __restrict__ wqrow = &Wq[(size_t)(16 * w + m) * D];
  const float* __restrict__ wkrow = &Wk[(size_t)(16 * w + m) * D];
  const float* arow = &xt[m * 1024];
  const int    stag = m * 4;                // row stagger for A reads

  v8f cq = {}; v8f ck = {};
  for (int kb = 0; kb < D; kb += 4) {
    if ((kb & 63) == 0) {                   // global_prefetch_b8 on W streams
      __builtin_prefetch(&wqrow[kb + 128], 0, 1);
      __builtin_prefetch(&wkrow[kb + 128], 0, 1);
    }
    v2f a   = *reinterpret_cast<const v2f*>(&arow[(kb + kof + stag) & 1023]);
    v2f bfq = *reinterpret_cast<const v2f*>(&wqrow[kb + kof]);
    v2f bfk = *reinterpret_cast<const v2f*>(&wkrow[kb + kof]);
    cq = __builtin_amdgcn_wmma_f32_16x16x4_f32(false, a, false, bfq,
                                               (short)0, cq, false, false);
    ck = __builtin_amdgcn_wmma_f32_16x16x4_f32(false, a, false, bfk,
                                               (short)0, ck, false, false);
  }

  const float biasq = bq[16 * w + m];
  const float biask = bk[16 * w + m];
  const int   mb    = (l >> 4) * 8;         // lanes 16-31 hold rows M=r+8
#pragma unroll
  for (int r = 0; r < 8; ++r) {
    const size_t o = (size_t)(bt0 + r + mb) * KQ + 16 * w + m;
    qout[o] = cq[r] + biasq;
    kout[o] = ck[r] + biask;
  }
}

// ---------------------------------------------------------------------------
// Kernel 2: fused similarity + running top-8 + gather.
// grid = B*(T/16) blocks, 256 threads.  Never materializes sim[B,T,T].
// q tile (shared A operand, 16x128) async-staged to LDS once, reused for all
// 16 source chunks by all 8 waves.  Per chunk: wave w computes sim sub-tile
// cols [16w,16w+16) with 32 f32 WMMAs (K=128); lanes 0..15 of wave 0 merge
// into per-row sorted top-8 (strict '>' => lowest index wins ties, matching
// jax.lax.top_k).  Gather streams x rows out with b128 copies (x[b] is 8 MB,
// L2-resident, so only the 537 MB of stores touch HBM).
// ---------------------------------------------------------------------------
__global__ __launch_bounds__(256) void sim_topk_gather_kernel(
    const float* __restrict__ x, const float* __restrict__ q,
    const float* __restrict__ k, float* __restrict__ gathered,
    float* __restrict__ idx_out, float* __restrict__ val_out)
{
  constexpr int QS = 132;                    // padded q-tile row stride
  constexpr int SS = 132;                    // padded sim row stride
  __shared__ float qt[16 * QS];
  __shared__ float sim[16 * SS];
  __shared__ int   pick[16 * TOPK];

  const int tid  = threadIdx.x;
  const int w    = tid >> 5;
  const int l    = tid & 31;
  const int m    = l & 15;
  const int kof  = (l >> 4) * 2;
  const int b    = blockIdx.x >> 7;          // 128 row-tiles per batch
  const int row0 = (blockIdx.x & 127) * 16;  // first q row of this tile
  const float scale = 0.08838834764831845f;  // 1/sqrt(128)

  // stage q tile [16 x 128] -> LDS (async, b128; 2 transfers per thread)
  for (int i = tid; i < 16 * KQ / 4; i += 256) {
    const int e = i * 4;
    const int r = e >> 7, c = e & 127;
    async_copy_b128(&q[((size_t)b * TT + row0 + r) * KQ + c], &qt[r * QS + c]);
  }
  async_join();
  __syncthreads();

  float tv[TOPK]; int ti[TOPK];
#pragma unroll
  for (int j = 0; j < TOPK; ++j) { tv[j] = -3.4e38f; ti[j] = 0; }

  for (int sc = 0; sc < TT; sc += 128) {
    // ---- 16x16 sim sub-tile for source cols [sc+16w, sc+16w+16) ----
    const size_t kbase = ((size_t)b * TT + sc + 16 * w + m) * KQ;
    __builtin_prefetch(&k[kbase + (size_t)128 * KQ], 0, 1);  // next chunk
    v8f acc = {};
    for (int kb = 0; kb < KQ; kb += 4) {
      v2f a  = *reinterpret_cast<const v2f*>(&qt[m * QS + kb + kof]);
      v2f bf = *reinterpret_cast<const v2f*>(&k[kbase + kb + kof]);
      acc = __builtin_amdgcn_wmma_f32_16x16x4_f32(false, a, false, bf,
                                                  (short)0, acc, false, false);
    }
    const int mb = (l >> 4) * 8;
#pragma unroll
    for (int r = 0; r < 8; ++r)
      sim[(r + mb) * SS + 16 * w + m] = acc[r] * scale;
    __syncthreads();

    // ---- merge chunk into running sorted top-8 (one lane per row) ----
    if (tid < 16) {
      for (int j = 0; j < 128; ++j) {
        float v = sim[tid * SS + j];
        int  id = sc + j;
#pragma unroll
        for (int t = 0; t < TOPK; ++t) {
          const bool sw = v > tv[t];
          const float nv = sw ? v : tv[t]; const int ni = sw ? id : ti[t];
          const float ov = sw ? tv[t] : v; const int oi = sw ? ti[t] : id;
          tv[t] = nv; ti[t] = ni; v = ov; id = oi;
        }
      }
    }
    __syncthreads();
  }

  // ---- emit indices (as float: tuple outputs share one float d_out) ----
  if (tid < 16) {
    const size_t obase = ((size_t)b * TT + row0 + tid) * TOPK;
#pragma unroll
    for (int j = 0; j < TOPK; ++j) {
      idx_out[obase + j] = (float)ti[j];
      val_out[obase + j] = tv[j];
      pick[tid * TOPK + j] = ti[j];
    }
  }
  __syncthreads();

  // ---- gather: 128 (row, slot) pairs, 16 per wave, float4 streams ----
  for (int p = 16 * w; p < 16 * w + 16; ++p) {
    const int row = p >> 3, jj = p & 7;
    const int src = pick[p];
    const float* __restrict__ s = &x[((size_t)b * TT + src) * D];
    float* __restrict__ d =
        &gathered[(((size_t)b * TT + row0 + row) * TOPK + jj) * D];
    for (int c = l * 4; c < D; c += 128)
      *reinterpret_cast<float4*>(&d[c]) =
          *reinterpret_cast<const float4*>(&s[c]);
  }
}

// ---------------------------------------------------------------------------
extern "C" void kernel_launch(void* const* d_in, const int* in_sizes, int n_in,
                              void* d_out, int out_size, void* d_ws,
                              size_t ws_size, hipStream_t stream) {
  (void)in_sizes; (void)n_in; (void)out_size; (void)ws_size;
  const float* x  = (const float*)d_in[0];
  const float* Wq = (const float*)d_in[1];
  const float* bq = (const float*)d_in[2];
  const float* Wk = (const float*)d_in[3];
  const float* bk = (const float*)d_in[4];
  // d_in[5] = k_topk scalar; fixed at 8 in this problem instance.

  float* qws = (float*)d_ws;                       // [B*T, 128]  (8 MB)
  float* kws = qws + (size_t)NB * TT * KQ;         // [B*T, 128]  (8 MB)

  float* gathered = (float*)d_out;                            // [B,T,8,D]
  float* idx_out  = gathered + (size_t)NB * TT * TOPK * D;    // [B,T,8]
  float* val_out  = idx_out  + (size_t)NB * TT * TOPK;        // [B,T,8]

  const int tiles = NB * TT / 16;   // 1024
  qk_proj_kernel<<<tiles, 256, 0, stream>>>(x, Wq, bq, Wk, bk, qws, kws);
  sim_topk_gather_kernel<<<tiles, 256, 0, stream>>>(x, qws, kws, gathered,
                                                    idx_out, val_out);
}